// RNN_43868795961910
// MI455X (gfx1250) — compile-verified
//
#include <hip/hip_runtime.h>
#include <hip/hip_bf16.h>

typedef float v2f __attribute__((ext_vector_type(2)));
typedef float v8f __attribute__((ext_vector_type(8)));

#define USE_ASYNC_X 1   // async global->LDS path for the x tile (gfx1250)

namespace {
constexpr int HID    = 50;
constexpr int IN0    = 12;
constexpr int TSTEPS = 512;
constexpr int MT     = 16;                 // batch rows per workgroup (one WMMA M-tile)
constexpr int NT     = 13;                 // ceil(4*HID / 16) N-tiles (N padded to 208)
constexpr int K0S    = 16;                 // layer0 k-steps: K = 50+12 -> pad 64
constexpr int K1S    = 26;                 // layer1 k-steps: K = 50+50 -> pad 104
constexpr int NPAD   = NT * 16;            // 208
constexpr int GST    = 212;                // gates LDS row stride (bank-skewed)
constexpr int CST    = 52;                 // cell-state LDS row stride
constexpr int THREADS = 256;
constexpr int WAVES   = THREADS / 32;

constexpr int SZ_B0 = NT * K0S * 64;       // fragment-major weights, layer0
constexpr int SZ_B1 = NT * K1S * 64;       // fragment-major weights, layer1
constexpr int SZ_A0 = (K0S * 4) * MT;      // [K=64][M=16] column-major (h0 | x_t)
constexpr int SZ_A1 = (K1S * 4) * MT;      // [K=104][M=16] column-major (h1 | h0_new)
constexpr int SZ_G  = MT * GST;
constexpr int SZ_C  = MT * CST;

constexpr int OFF_B0  = 0;
constexpr int OFF_B1  = OFF_B0 + SZ_B0;
constexpr int OFF_A0  = OFF_B1 + SZ_B1;
constexpr int OFF_A1  = OFF_A0 + SZ_A0;
constexpr int OFF_G0  = OFF_A1 + SZ_A1;
constexpr int OFF_G1  = OFF_G0 + SZ_G;
constexpr int OFF_C0  = OFF_G1 + SZ_G;
constexpr int OFF_C1  = OFF_C0 + SZ_C;
constexpr int OFF_BI0 = OFF_C1 + SZ_C;
constexpr int OFF_BI1 = OFF_BI0 + NPAD;
constexpr int SMEM_FLOATS = OFF_BI1 + NPAD;   // ~46.5k floats (~182 KB)
static_assert(SMEM_FLOATS * 4 < 320 * 1024, "must fit WGP LDS");
}

// Fast transcendentals: gfx1250 has v_tanh_f32 as a 1-op TRANS instruction.
__device__ __forceinline__ float fast_tanh(float v) {
#if __has_builtin(__builtin_amdgcn_tanhf)
  return __builtin_amdgcn_tanhf(v);
#elif __has_builtin(__builtin_amdgcn_tanh_f32)
  return __builtin_amdgcn_tanh_f32(v);
#else
  return tanhf(v);
#endif
}

__device__ __forceinline__ float fast_sigmoid(float v) {
  // sigmoid(x) = 0.5*tanh(x/2) + 0.5  (one TRANS op, no division)
  return __builtin_fmaf(fast_tanh(v * 0.5f), 0.5f, 0.5f);
}

__device__ __forceinline__ v8f wmma_f32_4(v2f a, v2f b, v8f c) {
#if __has_builtin(__builtin_amdgcn_wmma_f32_16x16x4_f32)
  // 8-arg form: (neg_a, A, neg_b, B, c_mod, C, reuse_a, reuse_b)
  return __builtin_amdgcn_wmma_f32_16x16x4_f32(false, a, false, b, (short)0, c,
                                               false, false);
#else
  c[0] += a.x * b.x + a.y * b.y;   // degenerate fallback (histogram shows wmma==0)
  return c;
#endif
}

// One layer's gate GEMM for this workgroup's 16-row batch tile.
// sA: [K][16] column-major activations, sB: fragment-major weights,
// sBias: combined (b_ih+b_hh) padded to NPAD, sG: activated gate output.
template <int KS>
__device__ __forceinline__ void lstm_gate_gemm(const float* __restrict__ sA,
                                               const float* __restrict__ sB,
                                               const float* __restrict__ sBias,
                                               float* __restrict__ sG,
                                               int wid, int lane) {
  const int ml   = lane & 15;
  const int kh   = (lane >> 4) << 1;   // 0 or 2: per-lane K sub-offset (A frag)
  const int mrow = (lane >> 4) << 3;   // 0 or 8: per-lane M offset (C/D frags)

  // A fragments for all k-steps (conflict-free b32 reads from column-major LDS)
  v2f a[KS];
#pragma unroll
  for (int ks = 0; ks < KS; ++ks) {
    const int k = ks * 4 + kh;
    a[ks].x = sA[k * MT + ml];
    a[ks].y = sA[(k + 1) * MT + ml];
  }

  for (int nt = wid; nt < NT; nt += WAVES) {
    const int n = nt * 16 + ml;
    const float bv = sBias[n];
    v8f acc = {bv, bv, bv, bv, bv, bv, bv, bv};
    const float* bp = sB + (nt * KS) * 64 + lane * 2;  // fragment-major: ds_load_b64
#pragma unroll
    for (int ks = 0; ks < KS; ++ks) {
      const v2f b = *(const v2f*)(bp + ks * 64);
      acc = wmma_f32_4(a[ks], b, acc);
    }
    if (n < 4 * HID) {
      // Branchless activation via tanh identity:
      //   'g' gate (tanh):     act = tanh(v)            -> mi=1.0, ao=0.0
      //   i/f/o  (sigmoid):    act = 0.5*tanh(v/2)+0.5  -> mi=0.5, ao=0.5
      const bool isg = (n >= 2 * HID) && (n < 3 * HID);
      const float mi = isg ? 1.0f : 0.5f;
      const float ao = isg ? 0.0f : 0.5f;
#pragma unroll
      for (int j = 0; j < 8; ++j) {
        const float t = fast_tanh(acc[j] * mi);
        sG[(mrow + j) * GST + n] = __builtin_fmaf(t, mi, ao);
      }
    }
  }
}

// Elementwise LSTM cell update; writes new h into next-step A buffers.
// Thread mapping is div/mod free: m = tid&15, h strides by 16.
__device__ __forceinline__ void lstm_cell_update(const float* __restrict__ sG,
                                                 float* __restrict__ sC,
                                                 float* __restrict__ sHrec,
                                                 float* __restrict__ sHnext,
                                                 int tid) {
  const int m = tid & 15;
#pragma unroll
  for (int h = tid >> 4; h < HID; h += THREADS / 16) {
    const float ig = sG[m * GST + h];
    const float fg = sG[m * GST + HID + h];
    const float gg = sG[m * GST + 2 * HID + h];
    const float og = sG[m * GST + 3 * HID + h];
    const float c = __builtin_fmaf(fg, sC[m * CST + h], ig * gg);
    sC[m * CST + h] = c;
    const float hn = og * fast_tanh(c);
    sHrec[h * MT + m] = hn;                       // recurrent input, cols 0..49
    if (sHnext) sHnext[(HID + h) * MT + m] = hn;  // next layer input, cols 50..99
  }
}

__global__ __launch_bounds__(THREADS) void lstm2_fused_kernel(
    const float* __restrict__ x,
    const float* __restrict__ w_ih0, const float* __restrict__ w_hh0,
    const float* __restrict__ b_ih0, const float* __restrict__ b_hh0,
    const float* __restrict__ w_ih1, const float* __restrict__ w_hh1,
    const float* __restrict__ b_ih1, const float* __restrict__ b_hh1,
    const float* __restrict__ w_out, const float* __restrict__ b_out,
    float* __restrict__ out) {
  extern __shared__ float smem[];
  float* sB0  = smem + OFF_B0;
  float* sB1  = smem + OFF_B1;
  float* sA0  = smem + OFF_A0;
  float* sA1  = smem + OFF_A1;
  float* sG0  = smem + OFF_G0;
  float* sG1  = smem + OFF_G1;
  float* sC0  = smem + OFF_C0;
  float* sC1  = smem + OFF_C1;
  float* sBI0 = smem + OFF_BI0;
  float* sBI1 = smem + OFF_BI1;

  const int tid  = threadIdx.x;
  const int lane = tid & 31;
  const int wid  = tid >> 5;

  // ---- one-time staging -------------------------------------------------
  for (int i = tid; i < SZ_A0; i += THREADS) sA0[i] = 0.0f;
  for (int i = tid; i < SZ_A1; i += THREADS) sA1[i] = 0.0f;
  for (int i = tid; i < SZ_C; i += THREADS) { sC0[i] = 0.0f; sC1[i] = 0.0f; }
  if (tid < NPAD) {
    sBI0[tid] = (tid < 4 * HID) ? (b_ih0[tid] + b_hh0[tid]) : 0.0f;
    sBI1[tid] = (tid < 4 * HID) ? (b_ih1[tid] + b_hh1[tid]) : 0.0f;
  }
  // layer0 weights -> fragment-major LDS: B[k][n] = (k<50 ? w_hh0[n][k] : w_ih0[n][k-50])
  for (int idx = tid; idx < SZ_B0; idx += THREADS) {
    const int f = idx >> 6, r = idx & 63;
    const int l2 = r >> 1, v = r & 1;
    const int nt = f / K0S, ks = f - nt * K0S;
    const int k = ks * 4 + ((l2 >> 4) << 1) + v;
    const int n = nt * 16 + (l2 & 15);
    float val = 0.0f;
    if (n < 4 * HID) {
      if (k < HID)            val = w_hh0[n * HID + k];
      else if (k < HID + IN0) val = w_ih0[n * IN0 + (k - HID)];
    }
    sB0[idx] = val;
  }
  // layer1 weights: B[k][n] = (k<50 ? w_hh1[n][k] : k<100 ? w_ih1[n][k-50] : 0)
  for (int idx = tid; idx < SZ_B1; idx += THREADS) {
    const int f = idx >> 6, r = idx & 63;
    const int l2 = r >> 1, v = r & 1;
    const int nt = f / K1S, ks = f - nt * K1S;
    const int k = ks * 4 + ((l2 >> 4) << 1) + v;
    const int n = nt * 16 + (l2 & 15);
    float val = 0.0f;
    if (n < 4 * HID) {
      if (k < HID)            val = w_hh1[n * HID + k];
      else if (k < 2 * HID)   val = w_ih1[n * HID + (k - HID)];
    }
    sB1[idx] = val;
  }
  __syncthreads();

  // ---- fused 2-layer recurrence over T ---------------------------------
  const long bbase = (long)blockIdx.x * MT;
  for (int t = 0; t < TSTEPS; ++t) {
    // x_t tile -> A0 columns 50..61 (column-major [K][16])
    if (tid < MT * IN0) {
      const int m = tid / IN0, i = tid - m * IN0;
      const long gx = (bbase + m) * (long)(TSTEPS * IN0) + (long)t * IN0 + i;
      const float* gptr = x + gx;
      float* dptr = &sA0[(HID + i) * MT + m];
#if defined(__gfx1250__) && USE_ASYNC_X
      // ASYNCcnt-tracked direct global->LDS copy (no VGPR bounce).
      // LDS byte address = low 32 bits of the generic pointer (flat->LDS
      // aperture truncation per ISA 10.2).
      const unsigned int ldsaddr = (unsigned int)(unsigned long long)dptr;
      asm volatile("global_load_async_to_lds_b32 %0, %1, off"
                   :
                   : "v"(ldsaddr), "v"(gptr)
                   : "memory");
#else
      *dptr = *gptr;
#endif
      if (t + 1 < TSTEPS) __builtin_prefetch(gptr + IN0, 0, 0);  // global_prefetch_b8
    }
#if defined(__gfx1250__) && USE_ASYNC_X
    asm volatile("s_wait_asynccnt 0" ::: "memory");  // async writes landed in LDS
#endif
    __syncthreads();

    lstm_gate_gemm<K0S>(sA0, sB0, sBI0, sG0, wid, lane);
    __syncthreads();

    lstm_cell_update(sG0, sC0, sA0, sA1, tid);   // h0 -> A0[0..49], A1[50..99]
    __syncthreads();

    lstm_gate_gemm<K1S>(sA1, sB1, sBI1, sG1, wid, lane);
    __syncthreads();

    lstm_cell_update(sG1, sC1, sA1, nullptr, tid);  // h1 -> A1[0..49]
    __syncthreads();
  }

  // ---- head: sigmoid(h1_last @ w_out^T + b_out) ------------------------
  if (tid < MT) {
    float acc = b_out[0];
#pragma unroll 1
    for (int h = 0; h < HID; ++h) acc += sA1[h * MT + tid] * w_out[h];
    out[bbase + tid] = fast_sigmoid(acc);
  }
}

extern "C" void kernel_launch(void* const* d_in, const int* in_sizes, int n_in,
                              void* d_out, int out_size, void* d_ws, size_t ws_size,
                              hipStream_t stream) {
  (void)in_sizes; (void)n_in; (void)d_ws; (void)ws_size;
  const float* x     = (const float*)d_in[0];
  const float* w_ih0 = (const float*)d_in[1];
  const float* w_hh0 = (const float*)d_in[2];
  const float* b_ih0 = (const float*)d_in[3];
  const float* b_hh0 = (const float*)d_in[4];
  const float* w_ih1 = (const float*)d_in[5];
  const float* w_hh1 = (const float*)d_in[6];
  const float* b_ih1 = (const float*)d_in[7];
  const float* b_hh1 = (const float*)d_in[8];
  const float* w_out = (const float*)d_in[9];
  const float* b_out = (const float*)d_in[10];
  float* out = (float*)d_out;

  const int B = out_size;            // [B, 1]
  const int blocks = B / MT;         // 2048/16 = 128 workgroups
  const size_t smem = (size_t)SMEM_FLOATS * sizeof(float);
  (void)hipFuncSetAttribute((const void*)lstm2_fused_kernel,
                            hipFuncAttributeMaxDynamicSharedMemorySize,
                            (int)smem);
  lstm2_fused_kernel<<<blocks, THREADS, smem, stream>>>(
      x, w_ih0, w_hh0, b_ih0, b_hh0, w_ih1, w_hh1, b_ih1, b_hh1, w_out, b_out,
      out);
}